// LTFGW_semirelaxed_90082644066819
// MI455X (gfx1250) — compile-verified
//
#include <hip/hip_runtime.h>
#include <hip/hip_bf16.h>

#define NN    10000
#define DEG   15
#define MSUB  16      // subgraph size = DEG + 1
#define FDIM  128
#define KT    10
#define NTT   10
#define KL    100     // KT * NTT
#define NITER_CG 5
#define NTILE 7       // ceil(KL / 16)

typedef float v2f __attribute__((ext_vector_type(2)));
typedef float v8f __attribute__((ext_vector_type(8)));

// ---------------------------------------------------------------------------
// Prep: C2 = 0.5*(templates + templates^T), B = C2*C2, f2[kl] = ||tf[kl]||^2
// ws layout (floats): [0,1000) C2, [1000,2000) B, [2000,2100) f2
// ---------------------------------------------------------------------------
__global__ void ltfgw_prep(const float* __restrict__ templates,
                           const float* __restrict__ tf,
                           float* __restrict__ ws) {
  int t = blockIdx.x * blockDim.x + threadIdx.x;
  if (t < KT * NTT * NTT) {
    int k = t / (NTT * NTT);
    int rc = t % (NTT * NTT);
    int r = rc / NTT, c = rc % NTT;
    float c2 = 0.5f * (templates[(k * NTT + r) * NTT + c] +
                       templates[(k * NTT + c) * NTT + r]);
    ws[t] = c2;
    ws[KT * NTT * NTT + t] = c2 * c2;
  }
  if (t < KL) {
    const float* p = tf + t * FDIM;
    float s = 0.f;
    for (int f = 0; f < FDIM; ++f) s += p[f] * p[f];
    ws[2 * KT * NTT * NTT + t] = s;
  }
}

// ---------------------------------------------------------------------------
// In-place  mat[16][KL] <- C1[16][16] @ mat   via V_WMMA_F32_16X16X4_F32.
// Per 16-column tile, all B-fragment reads happen before the D writes, so the
// update is safe in place. EXEC is all-ones (no divergence around the WMMA).
// ---------------------------------------------------------------------------
__device__ __forceinline__ void c1_apply(const float* __restrict__ C1b,
                                         float* mat, int row, int hi) {
  #pragma unroll
  for (int t = 0; t < NTILE; ++t) {
    int col = t * 16 + row;
    bool ok = (col < KL);
    v2f bf[4];
    #pragma unroll
    for (int s = 0; s < 4; ++s) {
      if (ok) {
        bf[s].x = mat[(4 * s + 2 * hi) * KL + col];
        bf[s].y = mat[(4 * s + 2 * hi + 1) * KL + col];
      } else {
        bf[s].x = 0.f; bf[s].y = 0.f;
      }
    }
    v8f d = {0.f, 0.f, 0.f, 0.f, 0.f, 0.f, 0.f, 0.f};
    #pragma unroll
    for (int s = 0; s < 4; ++s) {
      v2f a = *(const v2f*)(C1b + row * MSUB + 4 * s + 2 * hi);
      d = __builtin_amdgcn_wmma_f32_16x16x4_f32(false, a, false, bf[s],
                                                (short)0, d, false, false);
    }
    if (ok) {
      #pragma unroll
      for (int r = 0; r < 8; ++r) mat[(r + 8 * hi) * KL + col] = d[r];
    }
  }
}

// dst[b][k][m] = sum_l src[b][k][l] * C2[k][l][m]   (16*KL outputs, lane-strided)
__device__ __forceinline__ void mulC2(const float* __restrict__ src,
                                      float* __restrict__ dst,
                                      const float (*C2)[NTT][NTT], int lane) {
  for (int i = lane; i < MSUB * KL; i += 32) {
    int b = i / KL;
    int km = i % KL;
    int k = km / NTT, m = km % NTT;
    const float* Tr = src + b * KL + k * NTT;
    float s = 0.f;
    #pragma unroll
    for (int l = 0; l < NTT; ++l) s += Tr[l] * C2[k][l][m];
    dst[i] = s;
  }
}

// ---------------------------------------------------------------------------
// One wave (32 lanes) = one graph node. Whole srFGW solve in LDS.
// ---------------------------------------------------------------------------
__global__ void __launch_bounds__(32) ltfgw_main(
    const float* __restrict__ x, const int* __restrict__ ei,
    const float* __restrict__ tf, const float* __restrict__ alpha0,
    const float* __restrict__ ws, float* __restrict__ out) {
  __shared__ float sC2[KT][NTT][NTT];
  __shared__ float sB[KT][NTT][NTT];
  __shared__ float sf2[KL];
  __shared__ float sC1[MSUB][MSUB];
  __shared__ float sM[MSUB][KL];
  __shared__ float sT[MSUB][KL];
  __shared__ float sD[MSUB][KL];
  __shared__ float sCTC[MSUB][KL];
  __shared__ float sCDC[MSUB][KL];
  __shared__ int sIdx[MSUB];
  __shared__ unsigned sRm[MSUB];
  __shared__ float sRed[32];
  __shared__ float sX2[MSUB];
  __shared__ float sQ[KL];
  __shared__ float sDq[KL];
  __shared__ float sBq[KL];
  __shared__ float sS[KT];
  __shared__ float sConst;

  const int lane = threadIdx.x;
  const int row = lane & 15;
  const int hi = lane >> 4;
  const int node = blockIdx.x;
  const float P = 1.0f / (float)MSUB;
  const float T0 = P / (float)NTT;

  const float a0 = alpha0[0];
  const float alpha = 1.0f / (1.0f + __expf(-a0));
  const float oma = 1.0f - alpha;

  // template-derived constants into LDS
  for (int i = lane; i < KT * NTT * NTT; i += 32) {
    (&sC2[0][0][0])[i] = ws[i];
    (&sB[0][0][0])[i] = ws[KT * NTT * NTT + i];
  }
  for (int i = lane; i < KL; i += 32) sf2[i] = ws[2 * KT * NTT * NTT + i];

  // subgraph node indices: center + DEG neighbors (dst row of edge_index)
  if (lane < MSUB)
    sIdx[lane] = (lane == 0) ? node : ei[NN * DEG + node * DEG + (lane - 1)];
  __syncthreads();

  // C1 via membership:  C1[a][b] = 1 iff idx[b] in nbrs(idx[a]); symmetrized
  if (lane < MSUB) {
    int ga = sIdx[lane];
    unsigned rm = 0u;
    for (int j = 0; j < DEG; ++j) {
      int nb = ei[NN * DEG + ga * DEG + j];
      #pragma unroll
      for (int b = 0; b < MSUB; ++b)
        if (nb == sIdx[b]) rm |= (1u << b);
    }
    sRm[lane] = rm;
  }
  __syncthreads();
  if (lane < MSUB) {
    unsigned rma = sRm[lane];
    float rsum = 0.f;
    #pragma unroll
    for (int b = 0; b < MSUB; ++b) {
      unsigned bit = ((rma >> b) & 1u) | ((sRm[b] >> lane) & 1u);
      float v = (float)bit;
      sC1[lane][b] = v;
      rsum += v;
    }
    sRed[lane] = rsum;
  }
  __syncthreads();
  if (lane == 0) {
    float s = 0.f;
    for (int b = 0; b < MSUB; ++b) s += sRed[b];
    sConst = s * P * P;  // sum(C1*C1)*p*p  (C1 is 0/1)
  }
  __syncthreads();

  // Feature cost M[a][kl] = x2[a] + f2[kl] - 2 * feats @ tf^T, GEMM via WMMA.
  // A fragment streamed straight from global x (L2-resident gather).
  const float* gbase = x + (size_t)sIdx[row] * FDIM + 2 * hi;
  v8f acc[NTILE];
  #pragma unroll
  for (int t = 0; t < NTILE; ++t) acc[t] = (v8f){0.f,0.f,0.f,0.f,0.f,0.f,0.f,0.f};
  float sq = 0.f;
  for (int kk = 0; kk < FDIM; kk += 4) {
    v2f a = *(const v2f*)(gbase + kk);
    sq += a.x * a.x + a.y * a.y;
    #pragma unroll
    for (int t = 0; t < NTILE; ++t) {
      int col = t * 16 + row;
      v2f b = {0.f, 0.f};
      if (col < KL) b = *(const v2f*)(tf + col * FDIM + kk + 2 * hi);
      acc[t] = __builtin_amdgcn_wmma_f32_16x16x4_f32(false, a, false, b,
                                                     (short)0, acc[t],
                                                     false, false);
    }
  }
  sRed[lane] = sq;  // lane L holds half of ||feats[row]||^2; L and L+16 combine
  __syncthreads();
  if (lane < MSUB) sX2[lane] = sRed[lane] + sRed[lane + 16];
  __syncthreads();
  #pragma unroll
  for (int t = 0; t < NTILE; ++t) {
    int col = t * 16 + row;
    if (col < KL) {
      #pragma unroll
      for (int r = 0; r < 8; ++r)
        sM[r + 8 * hi][col] = sX2[r + 8 * hi] + sf2[col] - 2.0f * acc[t][r];
    }
  }

  // T0 = p / nt
  for (int i = lane; i < MSUB * KL; i += 32) (&sT[0][0])[i] = T0;
  __syncthreads();

  // ---- conditional-gradient iterations ----
  for (int it = 0; it < NITER_CG; ++it) {
    // q[kl] = sum_a T
    for (int i = lane; i < KL; i += 32) {
      float s = 0.f;
      #pragma unroll
      for (int a = 0; a < MSUB; ++a) s += sT[a][i];
      sQ[i] = s;
    }
    __syncthreads();
    // Bq[k][l] = sum_t B[k][l][t]*q[k][t]
    for (int i = lane; i < KL; i += 32) {
      int k = i / NTT, l = i % NTT;
      float s = 0.f;
      #pragma unroll
      for (int t = 0; t < NTT; ++t) s += sB[k][l][t] * sQ[k * NTT + t];
      sBq[i] = s;
    }
    // CTC = C1 @ (T @ C2)
    mulC2(&sT[0][0], &sCTC[0][0], sC2, lane);
    __syncthreads();
    c1_apply(&sC1[0][0], &sCTC[0][0], row, hi);
    __syncthreads();
    // G -> row-wise argmin -> D = p*onehot - T
    for (int i = lane; i < MSUB * KT; i += 32) {
      int a = i / KT, k = i % KT;
      float best = 3.4e38f;
      int bl = 0;
      #pragma unroll
      for (int l = 0; l < NTT; ++l) {
        int c = k * NTT + l;
        float g = oma * sM[a][c] + alpha * (2.f * sBq[c] - 4.f * sCTC[a][c]);
        if (g < best) { best = g; bl = l; }
      }
      #pragma unroll
      for (int l = 0; l < NTT; ++l) {
        int c = k * NTT + l;
        sD[a][c] = ((l == bl) ? P : 0.f) - sT[a][c];
      }
    }
    __syncthreads();
    // dq
    for (int i = lane; i < KL; i += 32) {
      float s = 0.f;
      #pragma unroll
      for (int a = 0; a < MSUB; ++a) s += sD[a][i];
      sDq[i] = s;
    }
    // CDC = C1 @ (D @ C2)
    mulC2(&sD[0][0], &sCDC[0][0], sC2, lane);
    __syncthreads();
    c1_apply(&sC1[0][0], &sCDC[0][0], row, hi);
    __syncthreads();
    // exact line search per template k (lanes 0..9)
    if (lane < KT) {
      int k = lane;
      float MD = 0.f, DCDC = 0.f, DCTC = 0.f, TCDC = 0.f;
      for (int a = 0; a < MSUB; ++a) {
        #pragma unroll
        for (int l = 0; l < NTT; ++l) {
          int c = k * NTT + l;
          float d = sD[a][c];
          MD += sM[a][c] * d;
          DCDC += d * sCDC[a][c];
          DCTC += d * sCTC[a][c];
          TCDC += sT[a][c] * sCDC[a][c];
        }
      }
      float dqBdq = 0.f, qBdq = 0.f;
      #pragma unroll
      for (int l = 0; l < NTT; ++l)
        #pragma unroll
        for (int t = 0; t < NTT; ++t) {
          float b = sB[k][l][t];
          dqBdq += sDq[k * NTT + l] * b * sDq[k * NTT + t];
          qBdq += sQ[k * NTT + l] * b * sDq[k * NTT + t];
        }
      float ac = alpha * (dqBdq - 2.f * DCDC);
      float bc = oma * MD + alpha * (2.f * qBdq - 2.f * (DCTC + TCDC));
      float s;
      if (ac > 0.f) {
        s = -bc / (2.f * ac + 1e-16f);
        s = fminf(fmaxf(s, 0.f), 1.f);
      } else {
        s = (ac + bc < 0.f) ? 1.f : 0.f;
      }
      sS[k] = s;
    }
    __syncthreads();
    // T += s[k] * D
    for (int i = lane; i < MSUB * KL; i += 32) {
      int k = (i % KL) / NTT;
      (&sT[0][0])[i] += sS[k] * (&sD[0][0])[i];
    }
    __syncthreads();
  }

  // ---- srFGW value at the fixed plan ----
  for (int i = lane; i < KL; i += 32) {
    float s = 0.f;
    #pragma unroll
    for (int a = 0; a < MSUB; ++a) s += sT[a][i];
    sQ[i] = s;
  }
  mulC2(&sT[0][0], &sCTC[0][0], sC2, lane);
  __syncthreads();
  c1_apply(&sC1[0][0], &sCTC[0][0], row, hi);
  __syncthreads();
  if (lane < KT) {
    int k = lane;
    float lin = 0.f, gw = 0.f;
    for (int a = 0; a < MSUB; ++a) {
      #pragma unroll
      for (int l = 0; l < NTT; ++l) {
        int c = k * NTT + l;
        lin += sM[a][c] * sT[a][c];
        gw += sT[a][c] * sCTC[a][c];
      }
    }
    float qBq = 0.f;
    #pragma unroll
    for (int l = 0; l < NTT; ++l)
      #pragma unroll
      for (int t = 0; t < NTT; ++t)
        qBq += sQ[k * NTT + l] * sB[k][l][t] * sQ[k * NTT + t];
    out[node * KT + k] = oma * lin + alpha * (sConst + qBq - 2.f * gw);
  }
}

extern "C" void kernel_launch(void* const* d_in, const int* in_sizes, int n_in,
                              void* d_out, int out_size, void* d_ws,
                              size_t ws_size, hipStream_t stream) {
  const float* x   = (const float*)d_in[0];
  const int*   ei  = (const int*)d_in[1];   // edge_index [2, N*DEG]
  const float* tpl = (const float*)d_in[2]; // templates [K,NT,NT]
  const float* tf  = (const float*)d_in[3]; // templates_features [K,NT,F]
  const float* a0  = (const float*)d_in[4]; // alpha0 [1]
  float* out = (float*)d_out;
  float* ws = (float*)d_ws;
  (void)in_sizes; (void)n_in; (void)out_size; (void)ws_size;

  ltfgw_prep<<<4, 256, 0, stream>>>(tpl, tf, ws);
  ltfgw_main<<<NN, 32, 0, stream>>>(x, ei, tf, a0, ws, out);
}